// KOP2d_55164559950404
// MI455X (gfx1250) — compile-verified
//
#include <hip/hip_runtime.h>

// ---------------------------------------------------------------------------
// KOP2d butterfly conv, B=32, C=128->128, H=W=64, log_n=6.
// Pipeline (all passes coalesced; transposes done through LDS):
//  1 k_xf_h : x -> H-butterfly(k11,dec)            -> T1 (img,h,w)
//  2 k_xf_w : T1 -> W-butterfly(k12,dec) + transpose -> Xf (p,b,i)
//  3 k_wf_h : pad(w) -> H-butterfly(kd1,dec)       -> T2c (img,h,w<3)  [sparse]
//  4 k_wf_w : T2c -> W-butterfly(kd2,dec)*64 + transpose -> Wf (p,i,o)
//  5 k_gemm : per-pixel complex GEMM, v_wmma_f32_16x16x4_f32,
//             A staged via global_load_async_to_lds_b128 (ASYNCcnt)
//  6 k_out_h: P -> H-butterfly(k21,inc) + transpose -> T1p (b,o,w,h)
//  7 k_out_w: T1p -> W-butterfly(k22,inc) -> Re -> out (b,o,h,w)
// ---------------------------------------------------------------------------

typedef __attribute__((ext_vector_type(2))) float v2f;
typedef __attribute__((ext_vector_type(8))) float v8f;

#define NPIX 4096
#define NB   32
#define NCH  128

__device__ __forceinline__ float2 cmul(float2 a, float2 b) {
    return make_float2(a.x * b.x - a.y * b.y, a.x * b.y + a.y * b.x);
}
__device__ __forceinline__ float2 cadd(float2 a, float2 b) {
    return make_float2(a.x + b.x, a.y + b.y);
}

// 6 butterfly stages over 64-point lines held in LDS.
// element(line, pos) = slab[line*lineStride + pos*posStride]
// tw: 6*32 complex 2x2 matrices, flat f32, (stage*32 + m)*8.
__device__ void bfly(float2* slab, int nlines, int lineStride, int posStride,
                     const float* tw, int increasing) {
    const int tid = threadIdx.x;
    const int total = nlines * 32;
    for (int idx = 0; idx < 6; ++idx) {
        const int log_s = increasing ? idx : 5 - idx;
        const int s = 1 << log_s;
        __syncthreads();
        for (int g = tid; g < total; g += blockDim.x) {
            const int line = g >> 5;
            const int m    = g & 31;            // = d*s + j
            const int j    = m & (s - 1);
            const int d    = m >> log_s;
            const int p0   = (d << (log_s + 1)) + j;
            const int p1   = p0 + s;
            const float* t = tw + (idx * 32 + m) * 8;
            const float2 t00 = make_float2(t[0], t[1]);
            const float2 t01 = make_float2(t[2], t[3]);
            const float2 t10 = make_float2(t[4], t[5]);
            const float2 t11 = make_float2(t[6], t[7]);
            float2* e0 = slab + line * lineStride + p0 * posStride;
            float2* e1 = slab + line * lineStride + p1 * posStride;
            const float2 x0 = *e0, x1 = *e1;
            *e0 = cadd(cmul(t00, x0), cmul(t01, x1));
            *e1 = cadd(cmul(t10, x0), cmul(t11, x1));
        }
    }
    __syncthreads();
}

// ---- 1: x H-pass, per image (b*128+i), natural layout in/out -------------
__global__ void k_xf_h(const float* __restrict__ x, const float* __restrict__ k11,
                       float* __restrict__ T1R, float* __restrict__ T1I) {
    __shared__ float2 img[64 * 65];
    __shared__ float tw[768];
    const int imgid = blockIdx.x;
    const int tid = threadIdx.x;
    for (int t = tid; t < 768; t += 256) tw[t] = k11[t];
    const float* src = x + (size_t)imgid * NPIX;
    for (int t = tid; t < NPIX; t += 256)
        img[(t >> 6) * 65 + (t & 63)] = make_float2(src[t], 0.f);
    bfly(img, 64, /*line=w*/1, /*pos=h*/65, tw, 0);
    float* dr = T1R + (size_t)imgid * NPIX;
    float* di = T1I + (size_t)imgid * NPIX;
    for (int t = tid; t < NPIX; t += 256) {
        const float2 v = img[(t >> 6) * 65 + (t & 63)];
        dr[t] = v.x; di[t] = v.y;
    }
}

// ---- 2: x W-pass + transpose to (p, b, i) --------------------------------
__global__ void k_xf_w(const float* __restrict__ T1R, const float* __restrict__ T1I,
                       const float* __restrict__ k12,
                       float* __restrict__ XfR, float* __restrict__ XfI) {
    __shared__ float2 slab[128 * 65];     // [i][w]
    __shared__ float tw[768];
    const int b = blockIdx.x >> 6;
    const int h = blockIdx.x & 63;
    const int tid = threadIdx.x;
    for (int t = tid; t < 768; t += 256) tw[t] = k12[t];
    for (int t = tid; t < 128 * 64; t += 256) {
        const int i = t >> 6, w = t & 63;
        const size_t g = ((size_t)(b * NCH + i)) * NPIX + h * 64 + w;
        slab[i * 65 + w] = make_float2(T1R[g], T1I[g]);
    }
    bfly(slab, 128, /*line=i*/65, /*pos=w*/1, tw, 0);
    for (int t = tid; t < 128 * 64; t += 256) {
        const int i = t & 127, w = t >> 7;
        const float2 v = slab[i * 65 + w];
        const size_t o = (((size_t)(h * 64 + w)) * NB + b) * NCH + i;
        XfR[o] = v.x; XfI[o] = v.y;
    }
}

// ---- 3: weight H-pass on padded 3x3 (only cols w<3 nonzero) --------------
__global__ void k_wf_h(const float* __restrict__ wgt, const float* __restrict__ kd1,
                       float* __restrict__ T2cR, float* __restrict__ T2cI) {
    __shared__ float2 wimg[64 * 4];       // (h, w<3), padded stride 4
    __shared__ float tw[768];
    const int imgid = blockIdx.x;         // o*128 + i
    const int tid = threadIdx.x;          // 96 threads
    for (int t = tid; t < 768; t += 96) tw[t] = kd1[t];
    for (int t = tid; t < 192; t += 96) {
        const int h = t / 3, w = t % 3;
        const float v = (h < 3) ? wgt[imgid * 9 + h * 3 + w] : 0.f;
        wimg[h * 4 + w] = make_float2(v, 0.f);
    }
    bfly(wimg, 3, /*line=w*/1, /*pos=h*/4, tw, 0);
    float* dr = T2cR + (size_t)imgid * 192;
    float* di = T2cI + (size_t)imgid * 192;
    for (int t = tid; t < 192; t += 96) {
        const float2 v = wimg[(t / 3) * 4 + (t % 3)];
        dr[t] = v.x; di[t] = v.y;
    }
}

// ---- 4: weight W-pass + transpose to (p, i, o), *64 ----------------------
__global__ void k_wf_w(const float* __restrict__ T2cR, const float* __restrict__ T2cI,
                       const float* __restrict__ kd2,
                       float* __restrict__ WfR, float* __restrict__ WfI) {
    __shared__ float2 slab[128 * 65];     // [o][w]
    __shared__ float tw[768];
    const int i = blockIdx.x >> 6;
    const int h = blockIdx.x & 63;
    const int tid = threadIdx.x;
    for (int t = tid; t < 768; t += 256) tw[t] = kd2[t];
    for (int t = tid; t < 128 * 64; t += 256) {
        const int o = t >> 6, w = t & 63;
        float2 v = make_float2(0.f, 0.f);
        if (w < 3) {
            const size_t g = (size_t)(o * NCH + i) * 192 + h * 3 + w;
            v = make_float2(T2cR[g], T2cI[g]);
        }
        slab[o * 65 + w] = v;
    }
    bfly(slab, 128, 65, 1, tw, 0);
    for (int t = tid; t < 128 * 64; t += 256) {
        const int o = t & 127, w = t >> 7;
        const float2 v = slab[o * 65 + w];
        const size_t d = (((size_t)(h * 64 + w)) * NCH + i) * NCH + o;
        WfR[d] = v.x * 64.f; WfI[d] = v.y * 64.f;   // * sqrt(H*W)
    }
}

// ---- 5: per-pixel complex GEMM via V_WMMA_F32_16X16X4_F32 ----------------
__global__ void __launch_bounds__(128)
k_gemm(const float* __restrict__ XfR, const float* __restrict__ XfI,
       const float* __restrict__ WfR, const float* __restrict__ WfI,
       float* __restrict__ PR, float* __restrict__ PI) {
    __shared__ float Ar[NB][NCH];
    __shared__ float Ai[NB][NCH];
    const int p   = blockIdx.x;
    const int tid = threadIdx.x;

    // Stage A (32x128 re+im) into LDS via CDNA5 async DMA loads (ASYNCcnt).
    {
        const float* xr = XfR + (size_t)p * (NB * NCH);
        const float* xi = XfI + (size_t)p * (NB * NCH);
        const unsigned lr = (unsigned)(size_t)(void*)&Ar[0][0];
        const unsigned li = (unsigned)(size_t)(void*)&Ai[0][0];
        #pragma unroll
        for (int c0 = 0; c0 < (NB * NCH) / 4; c0 += 128) {
            const int c = c0 + tid;
            asm volatile("global_load_async_to_lds_b128 %0, %1, off"
                         :: "v"(lr + (unsigned)c * 16u), "v"(xr + c * 4)
                         : "memory");
            asm volatile("global_load_async_to_lds_b128 %0, %1, off"
                         :: "v"(li + (unsigned)c * 16u), "v"(xi + c * 4)
                         : "memory");
        }
        asm volatile("s_wait_asynccnt 0x0" ::: "memory");
    }
    __syncthreads();

    const int lane  = tid & 31;
    const int wv    = tid >> 5;           // wave -> output cols [wv*32, wv*32+32)
    const int lrow  = lane & 15;
    const int khalf = (lane >> 4) << 1;   // A/B fragment K split across half-waves
    const size_t wbase = (size_t)p * (NCH * NCH);

    v8f accR[2][2] = {};
    v8f accI[2][2] = {};

    #pragma unroll 4
    for (int k0 = 0; k0 < NCH; k0 += 4) {
        const int kk = k0 + khalf;
        v2f arf[2], aif[2], ain[2];
        #pragma unroll
        for (int m = 0; m < 2; ++m) {
            const int row = m * 16 + lrow;
            v2f a; a.x = Ar[row][kk]; a.y = Ar[row][kk + 1]; arf[m] = a;
            v2f b; b.x = Ai[row][kk]; b.y = Ai[row][kk + 1]; aif[m] = b;
            ain[m] = -aif[m];             // f32 WMMA NEG covers C only
        }
        #pragma unroll
        for (int nt = 0; nt < 2; ++nt) {
            const int n = wv * 32 + nt * 16 + lrow;
            const float* br = WfR + wbase + (size_t)kk * NCH + n;
            const float* bi = WfI + wbase + (size_t)kk * NCH + n;
            if (k0 + 4 < NCH) {
                __builtin_prefetch(br + 4 * NCH, 0, 3);
                __builtin_prefetch(bi + 4 * NCH, 0, 3);
            }
            v2f brf; brf.x = br[0]; brf.y = br[NCH];
            v2f bif; bif.x = bi[0]; bif.y = bi[NCH];
            #pragma unroll
            for (int m = 0; m < 2; ++m) {
                accR[m][nt] = __builtin_amdgcn_wmma_f32_16x16x4_f32(
                    false, arf[m], false, brf, (short)0, accR[m][nt], false, false);
                accR[m][nt] = __builtin_amdgcn_wmma_f32_16x16x4_f32(
                    false, ain[m], false, bif, (short)0, accR[m][nt], false, false);
                accI[m][nt] = __builtin_amdgcn_wmma_f32_16x16x4_f32(
                    false, arf[m], false, bif, (short)0, accI[m][nt], false, false);
                accI[m][nt] = __builtin_amdgcn_wmma_f32_16x16x4_f32(
                    false, aif[m], false, brf, (short)0, accI[m][nt], false, false);
            }
        }
    }

    const int mofs = (lane >> 4) << 3;    // C/D: VGPR r -> M = r + 8*(lane>=16)
    #pragma unroll
    for (int m = 0; m < 2; ++m)
        #pragma unroll
        for (int nt = 0; nt < 2; ++nt)
            #pragma unroll
            for (int r = 0; r < 8; ++r) {
                const int M = m * 16 + r + mofs;
                const int N = wv * 32 + nt * 16 + lrow;
                const size_t o = ((size_t)p * NB + M) * NCH + N;
                PR[o] = accR[m][nt][r];
                PI[o] = accI[m][nt][r];
            }
}

// ---- 6: P H-pass (inc, k21) + transpose to (b, o, w, h) ------------------
__global__ void k_out_h(const float* __restrict__ PR, const float* __restrict__ PI,
                        const float* __restrict__ k21,
                        float* __restrict__ T1pR, float* __restrict__ T1pI) {
    __shared__ float2 slab[128 * 65];     // [o][h]
    __shared__ float tw[768];
    const int b = blockIdx.x >> 6;
    const int w = blockIdx.x & 63;
    const int tid = threadIdx.x;
    for (int t = tid; t < 768; t += 256) tw[t] = k21[t];
    for (int t = tid; t < 128 * 64; t += 256) {
        const int o = t & 127, h = t >> 7;
        const size_t g = (((size_t)(h * 64 + w)) * NB + b) * NCH + o;
        slab[o * 65 + h] = make_float2(PR[g], PI[g]);
    }
    bfly(slab, 128, 65, 1, tw, 1);        // along h, increasing
    for (int t = tid; t < 128 * 64; t += 256) {
        const int h = t & 63, o = t >> 6;
        const float2 v = slab[o * 65 + h];
        const size_t d = ((size_t)(b * NCH + o) * 64 + w) * 64 + h;
        T1pR[d] = v.x; T1pI[d] = v.y;
    }
}

// ---- 7: P W-pass (inc, k22), real part -> out ----------------------------
__global__ void k_out_w(const float* __restrict__ T1pR, const float* __restrict__ T1pI,
                        const float* __restrict__ k22, float* __restrict__ out) {
    __shared__ float2 img[64 * 65];       // [w][h]
    __shared__ float tw[768];
    const int imgid = blockIdx.x;         // b*128 + o
    const int tid = threadIdx.x;
    for (int t = tid; t < 768; t += 256) tw[t] = k22[t];
    const float* sr = T1pR + (size_t)imgid * NPIX;
    const float* si = T1pI + (size_t)imgid * NPIX;
    for (int t = tid; t < NPIX; t += 256)       // t = w*64 + h
        img[(t >> 6) * 65 + (t & 63)] = make_float2(sr[t], si[t]);
    bfly(img, 64, /*line=h*/1, /*pos=w*/65, tw, 1);
    float* dst = out + (size_t)imgid * NPIX;
    for (int t = tid; t < NPIX; t += 256) {     // t = h*64 + w
        const int h = t >> 6, w = t & 63;
        dst[t] = img[w * 65 + h].x;
    }
}

// ---------------------------------------------------------------------------
extern "C" void kernel_launch(void* const* d_in, const int* in_sizes, int n_in,
                              void* d_out, int out_size, void* d_ws, size_t ws_size,
                              hipStream_t stream) {
    const float* x   = (const float*)d_in[0];
    const float* wgt = (const float*)d_in[1];
    const float* kd1 = (const float*)d_in[2];
    const float* kd2 = (const float*)d_in[3];
    const float* k11 = (const float*)d_in[4];
    const float* k12 = (const float*)d_in[5];
    const float* k21 = (const float*)d_in[6];
    const float* k22 = (const float*)d_in[7];
    float* out = (float*)d_out;
    (void)in_sizes; (void)n_in; (void)out_size; (void)ws_size;

    const size_t nX = (size_t)NPIX * NB * NCH;    // 16,777,216
    const size_t nW = (size_t)NPIX * NCH * NCH;   // 67,108,864

    // Region A (2*nX): T1 [k1,k2] -> T2c [k3,k4] -> P [k5,k6]
    float* A = (float*)d_ws;
    float* C = A + 2 * nX;                        // Region C (2*nX): Xf [k2,k5]
    float* Bg = C + 2 * nX;                       // Region B (2*nW): Wf [k4,k5] -> T1p [k6,k7]

    float* T1R  = A;   float* T1I  = A + nX;
    float* T2cR = A;   float* T2cI = A + (size_t)16384 * 192;
    float* PRm  = A;   float* PIm  = A + nX;
    float* XfR  = C;   float* XfI  = C + nX;
    float* WfR  = Bg;  float* WfI  = Bg + nW;
    float* T1pR = Bg;  float* T1pI = Bg + nX;

    k_xf_h <<<dim3(NB * NCH),  dim3(256), 0, stream>>>(x, k11, T1R, T1I);
    k_xf_w <<<dim3(NB * 64),   dim3(256), 0, stream>>>(T1R, T1I, k12, XfR, XfI);
    k_wf_h <<<dim3(NCH * NCH), dim3(96),  0, stream>>>(wgt, kd1, T2cR, T2cI);
    k_wf_w <<<dim3(NCH * 64),  dim3(256), 0, stream>>>(T2cR, T2cI, kd2, WfR, WfI);
    k_gemm <<<dim3(NPIX),      dim3(128), 0, stream>>>(XfR, XfI, WfR, WfI, PRm, PIm);
    k_out_h<<<dim3(NB * 64),   dim3(256), 0, stream>>>(PRm, PIm, k21, T1pR, T1pI);
    k_out_w<<<dim3(NB * NCH),  dim3(256), 0, stream>>>(T1pR, T1pI, k22, out);
}